// Memory_21947282882926
// MI455X (gfx1250) — compile-verified
//
#include <hip/hip_runtime.h>
#include <hip/hip_bf16.h>

// ---------------- problem constants ----------------
#define B_      32
#define SEQ_    512
#define DIN_    512
#define IFACE_  919
#define NWORDS_ 1024
#define WSZ_    128
#define RHEADS_ 4
#define SCHUNKS 8           // SEQ split for the reduction pass

// output region offsets (floats) in d_out, concatenated tuple order:
// out, memory, link, usage, prec, ww, rw
#define OFF_OUT    0
#define OFF_MEM    16384
#define OFF_LINK   4210688
#define OFF_USAGE  37765120
#define OFF_PREC   37797888
#define OFF_WW     37830656
#define OFF_RW     37863424

// workspace layout (floats)
#define WS_XM    0            // 32*512        = 16384
#define WS_IF    16384        // 32*919        = 29408
#define WS_ER    45792        // 32*128        = 4096
#define WS_WV    49888        // 32*128        = 4096
#define WS_WWA   53984        // 32*1024       = 32768
#define WS_U     86752        // 32
#define WS_RM6   86784        // 32*4  (1e-6 * read_modes1)
#define WS_RWV   86912        // 32*4  (read_modes1 / 1024)
#define WS_PART  87040        // 32*8*512      = 131072 (partial SEQ sums)
// total: 218112 floats = 852 KB

typedef __attribute__((ext_vector_type(2))) float v2f;
typedef __attribute__((ext_vector_type(4))) float v4f;
typedef __attribute__((ext_vector_type(8))) float v8f;

__device__ __forceinline__ float sigm(float x) { return 1.0f / (1.0f + expf(-x)); }

// ---------------- kernel 1a: partial sums over SEQ chunks (b128 streaming reads) ----
__global__ __launch_bounds__(128) void k_reduce_part(const v4f* __restrict__ x4,
                                                     v4f* __restrict__ part4) {
  const int b  = blockIdx.x;            // 0..31
  const int c  = blockIdx.y;            // 0..7
  const int d4 = threadIdx.x;           // 0..127 (512/4 vector columns)
  const int rows = SEQ_ / SCHUNKS;      // 64
  const v4f* p = x4 + (size_t)b * SEQ_ * (DIN_ / 4)
                    + (size_t)c * rows * (DIN_ / 4) + d4;
  v4f s = {0.f, 0.f, 0.f, 0.f};
  #pragma unroll 4
  for (int i = 0; i < rows; ++i) s += p[(size_t)i * (DIN_ / 4)];
  part4[((size_t)b * SCHUNKS + c) * (DIN_ / 4) + d4] = s;
}

// ---------------- kernel 1b: combine partials -> xm[b,d] = mean_s x[b,s,d] ----------
__global__ __launch_bounds__(128) void k_reduce_comb(const v4f* __restrict__ part4,
                                                     v4f* __restrict__ xm4) {
  const int b  = blockIdx.x;            // 0..31
  const int d4 = threadIdx.x;           // 0..127
  v4f s = {0.f, 0.f, 0.f, 0.f};
  #pragma unroll
  for (int c = 0; c < SCHUNKS; ++c)
    s += part4[((size_t)b * SCHUNKS + c) * (DIN_ / 4) + d4];
  s *= (1.0f / (float)SEQ_);
  xm4[(size_t)b * (DIN_ / 4) + d4] = s;
}

// ---------------- kernel 2: iface = xm @ W via V_WMMA_F32_16X16X4_F32 ----------------
__global__ __launch_bounds__(32) void k_gemm_iface(const float* __restrict__ xm,
                                                   const float* __restrict__ Wm,
                                                   float* __restrict__ iface) {
  const int ntile = blockIdx.x;          // 0..57  (58*16 = 928 >= 919)
  const int mtile = blockIdx.y;          // 0..1
  const int lane  = threadIdx.x;         // 0..31
  const int half  = lane >> 4;           // K-pair select (A/B layout, 16x4 f32)
  const int l16   = lane & 15;
  const int col   = ntile * 16 + l16;
  const bool colok = (col < IFACE_);
  const int colc  = colok ? col : 0;     // clamp to stay in-bounds, mask later

  v8f acc = {};
  const float* arow = xm + (mtile * 16 + l16) * DIN_ + half * 2;
  for (int k0 = 0; k0 < DIN_; k0 += 4) {
    v2f a, b;
    a.x = arow[k0];
    a.y = arow[k0 + 1];
    const int krow = k0 + half * 2;
    float b0 = Wm[(size_t)krow * IFACE_ + colc];
    float b1 = Wm[(size_t)(krow + 1) * IFACE_ + colc];
    b.x = colok ? b0 : 0.0f;
    b.y = colok ? b1 : 0.0f;
    // D = A(16x4,f32) * B(4x16,f32) + C ; 8 args: neg_a,A,neg_b,B,c_mod,C,reuse_a,reuse_b
    acc = __builtin_amdgcn_wmma_f32_16x16x4_f32(false, a, false, b,
                                                (short)0, acc, false, false);
  }
  if (colok) {
    #pragma unroll
    for (int v = 0; v < 8; ++v) {
      const int row = half * 8 + v;      // C/D layout: vgpr v, lanes16-31 -> M=v+8
      iface[(size_t)(mtile * 16 + row) * IFACE_ + col] = acc[v];
    }
  }
}

// ---------------- kernel 3: per-batch derived quantities ----------------
__global__ __launch_bounds__(128) void k_derive(const float* __restrict__ iface,
                                                float* __restrict__ ws) {
  const int b = blockIdx.x;              // 0..31
  const int t = threadIdx.x;             // 0..127
  const float* f = iface + (size_t)b * IFACE_;
  // erase = sigmoid(iface[645:773]); write_vector = iface[773:901]
  ws[WS_ER + b * WSZ_ + t] = sigm(f[645 + t]);
  ws[WS_WV + b * WSZ_ + t] = f[773 + t];
  if (t == 0) {
    float fg[RHEADS_], rm1[RHEADS_], rwv[RHEADS_];
    for (int r = 0; r < RHEADS_; ++r) fg[r] = sigm(f[901 + r]);
    const float ag = sigm(f[905]);
    const float wg = sigm(f[906]);
    float ret = 1.0f;
    for (int r = 0; r < RHEADS_; ++r) {
      // softmax over the 3 read modes for head r; only mode 1 survives
      const float m0 = f[907 + r], m1 = f[911 + r], m2 = f[915 + r];
      float mx = fmaxf(m0, fmaxf(m1, m2));
      const float e0 = expf(m0 - mx), e1 = expf(m1 - mx), e2 = expf(m2 - mx);
      rm1[r] = e1 / (e0 + e1 + e2);
      rwv[r] = rm1[r] * (1.0f / (float)NWORDS_);
      ret *= (1.0f - fg[r] * rwv[r]);
      ws[WS_RM6 + b * RHEADS_ + r] = 1e-6f * rm1[r];
      ws[WS_RWV + b * RHEADS_ + r] = rwv[r];
    }
    const float u = 1e-6f * ret;         // usage value (constant over n)
    ws[WS_U + b] = u;
    const float wwc = wg * (1.0f - ag) * (1.0f / (float)NWORDS_);
    const float wga = wg * ag;
    float pw = 1.0f;                     // u^n, sequential cumprod like the reference
    for (int n = 0; n < NWORDS_; ++n) {
      const float alloc = (1.0f - u) * pw;
      ws[WS_WWA + b * NWORDS_ + n] = wga * alloc + wwc;
      pw *= u;
    }
  }
}

// ---------------- kernel 4: memory[b,n,w] (b128 NT streaming stores) --------------
__global__ __launch_bounds__(256) void k_fill_memory(const float* __restrict__ ws,
                                                     v4f* __restrict__ mem4) {
  const int i  = blockIdx.x * 256 + threadIdx.x;  // 0 .. 1048575 (v4f units)
  const int w4 = i & 31;                          // v4f group within 128-word
  const int n  = (i >> 5) & (NWORDS_ - 1);
  const int b  = i >> 15;
  const float wwn = ws[WS_WWA + b * NWORDS_ + n];
  const v4f er = ((const v4f*)(ws + WS_ER))[b * 32 + w4];
  const v4f wv = ((const v4f*)(ws + WS_WV))[b * 32 + w4];
  const v4f o = 1e-6f * (1.0f - wwn * er) + wwn * wv;
  __builtin_nontemporal_store(o, &mem4[i]);
}

// ---------------- kernel 5: link = 0 (33.5M floats, b128 NT stores) ----------------
__global__ __launch_bounds__(256) void k_zero_link(v4f* __restrict__ link4) {
  const int i = blockIdx.x * 256 + threadIdx.x;   // 0 .. 8388607
  const v4f z = {0.f, 0.f, 0.f, 0.f};
  __builtin_nontemporal_store(z, &link4[i]);
}

// ---------------- kernel 6: out, usage, prec, ww, rw ----------------
__global__ __launch_bounds__(256) void k_fill_small(const float* __restrict__ ws,
                                                    float* __restrict__ dout) {
  int i = blockIdx.x * 256 + threadIdx.x;         // 0 .. 245759
  if (i < 16384) {                                 // out[b, w*4+r] = 1e-6*rm1[b,r]
    const int b = i >> 9, r = i & 3;
    dout[OFF_OUT + i] = ws[WS_RM6 + b * RHEADS_ + r];
    return;
  }
  i -= 16384;
  if (i < 32768) {                                 // usage[b,n] = u[b]
    dout[OFF_USAGE + i] = ws[WS_U + (i >> 10)];
    return;
  }
  i -= 32768;
  if (i < 32768) {                                 // prec = ww
    dout[OFF_PREC + i] = ws[WS_WWA + i];
    return;
  }
  i -= 32768;
  if (i < 32768) {                                 // ww
    dout[OFF_WW + i] = ws[WS_WWA + i];
    return;
  }
  i -= 32768;
  {                                                // rw[b,n,r] = rm1[b,r]/1024
    const int b = i >> 12, r = i & 3;
    dout[OFF_RW + i] = ws[WS_RWV + b * RHEADS_ + r];
  }
}

extern "C" void kernel_launch(void* const* d_in, const int* in_sizes, int n_in,
                              void* d_out, int out_size, void* d_ws, size_t ws_size,
                              hipStream_t stream) {
  const float* x  = (const float*)d_in[0];   // (32, 512, 512)
  const float* Wm = (const float*)d_in[1];   // (512, 919)
  float* out = (float*)d_out;
  float* ws  = (float*)d_ws;

  // 1) xm = mean over SEQ of x  (two-pass: 256 blocks streaming, then combine)
  dim3 rgrid(B_, SCHUNKS);
  k_reduce_part<<<rgrid, 128, 0, stream>>>((const v4f*)x,
                                           (v4f*)(ws + WS_PART));
  k_reduce_comb<<<B_, 128, 0, stream>>>((const v4f*)(ws + WS_PART),
                                        (v4f*)(ws + WS_XM));

  // 2) iface = xm @ W  (WMMA f32 16x16x4)
  dim3 ggrid((IFACE_ + 15) / 16, B_ / 16);   // 58 x 2 waves
  k_gemm_iface<<<ggrid, 32, 0, stream>>>(ws + WS_XM, Wm, ws + WS_IF);

  // 3) per-batch derived state
  k_derive<<<B_, 128, 0, stream>>>(ws + WS_IF, ws);

  // 4) memory output (4.19M elems -> 1.05M b128 NT stores)
  k_fill_memory<<<(B_ * NWORDS_ * WSZ_ / 4) / 256, 256, 0, stream>>>(
      ws, (v4f*)(out + OFF_MEM));

  // 5) link output = 0 (33.55M elems -> 8.39M b128 NT stores)
  k_zero_link<<<(B_ * NWORDS_ * NWORDS_ / 4) / 256, 256, 0, stream>>>(
      (v4f*)(out + OFF_LINK));

  // 6) out / usage / prec / ww / rw (245760 elems)
  k_fill_small<<<245760 / 256, 256, 0, stream>>>(ws, out);
}